// PACKDETPostProcessor_71665824301802
// MI455X (gfx1250) — compile-verified
//
#include <hip/hip_runtime.h>
#include <stdint.h>

typedef unsigned int uint;

#define NIMG   8
#define NCLS   80
#define HWN    40000
#define CH     160            // hw elements per tile chunk
#define NCHUNK 250            // 40000 / 160
#define TOPK   1000
#define CAP    4096
#define OUTN   100
#define NMS_T  0.6f
#define LOGIT_T (-2.9444389791664403f)   // log(0.05/0.95): p>0.05 <=> logit > this

typedef uint  u32x4 __attribute__((ext_vector_type(4)));
typedef int   i32x8 __attribute__((ext_vector_type(8)));
typedef int   i32x4 __attribute__((ext_vector_type(4)));
typedef float v2f   __attribute__((ext_vector_type(2)));
typedef float v8f   __attribute__((ext_vector_type(8)));

// ---- workspace layout (uint-word offsets) ----
constexpr int HISTW = 0;                          // 3 levels * 8 imgs * 256 bins
constexpr int CNTW  = HISTW + 3 * NIMG * 256;     // 8
constexpr int CTXW  = CNTW + NIMG;                // 8 (selected radix prefix)
constexpr int REMW  = CTXW + NIMG;                // 8 (remaining count)
constexpr int CANDW = REMW + NIMG;                // uint2 cand[8][4096]
constexpr int TKEYW = CANDW + NIMG * CAP * 2;
constexpr int TIDXW = TKEYW + NIMG * TOPK;
constexpr int BOXW  = TIDXW + NIMG * TOPK;        // float[8][1000][4]
constexpr int OBOXW = BOXW + NIMG * TOPK * 4;     // offset boxes
constexpr int LBLW  = OBOXW + NIMG * TOPK * 4;
constexpr int VALW  = LBLW + NIMG * TOPK;
constexpr int ADJW  = VALW + NIMG * TOPK;         // u32[8][1000][32]
constexpr size_t ZERO_BYTES = (size_t)CANDW * 4;  // hists + counters + ctx + rem

__device__ __forceinline__ float sig(float x) { return 1.0f / (1.0f + __expf(-x)); }

// ---- Tensor Data Mover: 2D tile (tile_x elems wide, tile_y rows, row stride in elems), 4B elems ----
__device__ __forceinline__ void tdm_load_2d(uint lds_off, const void* g,
                                            uint tile_x, uint tile_y, uint stride_elems) {
    uint64_t ga = (uint64_t)(uintptr_t)g;
    u32x4 g0;
    g0.x = 1u;                                              // count=1, user descriptor
    g0.y = lds_off;                                         // lds_addr
    g0.z = (uint)ga;                                        // global_addr[31:0]
    g0.w = ((uint)(ga >> 32) & 0x01FFFFFFu) | (2u << 30);   // global_addr[56:32] | type=2
    uint td0 = stride_elems;   // tensor_dim0 (in-bounds by construction)
    uint td1 = tile_y;         // tensor_dim1
    i32x8 g1;
    g1[0] = (int)(2u << 16);                                            // data_size=4B, no multicast
    g1[1] = (int)((td0 & 0xFFFFu) << 16);                               // abar=0 | tensor_dim0[15:0]
    g1[2] = (int)(((td0 >> 16) & 0xFFFFu) | ((td1 & 0xFFFFu) << 16));   // tdim0 hi | tdim1 lo
    g1[3] = (int)(((td1 >> 16) & 0xFFFFu) | ((tile_x & 0xFFFFu) << 16));// tdim1 hi | tile_dim0
    g1[4] = (int)(tile_y & 0xFFFFu);                                    // tile_dim1 | tile_dim2=0
    g1[5] = (int)stride_elems;                                          // tensor_dim0_stride lo
    g1[6] = 0;                                                          // stride hi | dim1_stride lo
    g1[7] = 0;
    i32x4 z4 = {0, 0, 0, 0};
    i32x8 z8 = {0, 0, 0, 0, 0, 0, 0, 0};
    // amdgpu-toolchain (clang-23) 6-arg form: (g0, g1, g2, g3, g_ext, cpol)
    __builtin_amdgcn_tensor_load_to_lds(g0, g1, z4, z4, z8, 0);
}

// ---- streaming pass over box_cls: MODE 0/1/2 = radix histograms, MODE 3 = compaction ----
template <int MODE>
__global__ __launch_bounds__(256) void scanK(const float* __restrict__ cls,
                                             const float* __restrict__ cent,
                                             uint* __restrict__ W) {
    __shared__ float tile[NCLS * CH];   // 50 KB TDM destination
    __shared__ float ctl[CH];
    __shared__ uint  hl[256];
    const int n   = blockIdx.y;
    const int hwb = blockIdx.x * CH;
    const int tid = threadIdx.x;

    if (tid < 256) hl[tid] = 0;
    if (tid < CH)  ctl[tid] = sig(cent[n * HWN + hwb + tid]);

    const float* gsrc = cls + (size_t)n * NCLS * HWN + hwb;
    __builtin_prefetch(gsrc, 0, 1);
    if (tid == 0) {
        uint lds_off = (uint)(uintptr_t)(&tile[0]);
        tdm_load_2d(lds_off, gsrc, CH, NCLS, HWN);
    }
    __builtin_amdgcn_s_wait_tensorcnt(0);
    __syncthreads();

    uint ctx = 0;
    if (MODE >= 1) ctx = W[CTXW + n];
    uint2* cand = (uint2*)(W + CANDW);

    for (int e = tid; e < NCLS * CH; e += 256) {
        float v = tile[e];
        if (v > LOGIT_T) {
            int c   = e / CH;
            int hwl = e - c * CH;
            float p = sig(v);
            uint key = __float_as_uint(p * ctl[hwl]);
            if (MODE == 0) {
                atomicAdd(&hl[key >> 24], 1u);
            } else if (MODE == 1) {
                if ((key >> 24) == ctx) atomicAdd(&hl[(key >> 16) & 255u], 1u);
            } else if (MODE == 2) {
                if ((key >> 16) == ctx) atomicAdd(&hl[(key >> 8) & 255u], 1u);
            } else {
                if ((key >> 8) >= ctx) {
                    uint pos = atomicAdd(&W[CNTW + n], 1u);
                    if (pos < CAP) {
                        uint fe = (uint)(hwb + hwl) * NCLS + (uint)c;
                        cand[(size_t)n * CAP + pos] = make_uint2(key, fe);
                    }
                }
            }
        }
    }
    if (MODE < 3) {
        __syncthreads();
        if (tid < 256) {
            uint v = hl[tid];
            if (v) atomicAdd(&W[HISTW + MODE * NIMG * 256 + n * 256 + tid], v);
        }
    }
}

// ---- radix pick: suffix-scan of 256-bin histogram via triangular-matrix WMMA, then select ----
template <int LVL>
__global__ __launch_bounds__(32) void pickK(uint* __restrict__ W) {
    __shared__ float hf[256];
    __shared__ uint  hu[256];
    __shared__ float a2f[256];
    __shared__ float abv[256];
    const int n = blockIdx.x;
    const int lane = threadIdx.x;
    const uint* hist = W + HISTW + LVL * NIMG * 256 + n * 256;
    for (int q = 0; q < 8; ++q) {
        int b = q * 32 + lane;
        uint h = hist[b];
        hu[b] = h;
        hf[b] = (float)h;
    }
    __syncthreads();

#if __has_builtin(__builtin_amdgcn_wmma_f32_16x16x4_f32)
    // above[r*16+c] = sum_{c'>c} Hm[r][c'] + sum_{r'>r} rowsum[r']  as chained 16x16x4 WMMAs
    const int half = lane >> 4;
    const int lm   = lane & 15;
    v8f acc;  for (int v = 0; v < 8; ++v) acc[v] = 0.f;
    v8f acc2; for (int v = 0; v < 8; ++v) acc2[v] = 0.f;
#pragma unroll
    for (int k = 0; k < 4; ++k) {   // S1 = Hm x T,  T[c][j] = (c > j)
        int kr = 4 * k + (half ? 2 : 0);
        v2f a; a.x = hf[lm * 16 + kr]; a.y = hf[lm * 16 + kr + 1];
        v2f b; b.x = (kr > lm) ? 1.f : 0.f; b.y = (kr + 1 > lm) ? 1.f : 0.f;
        acc = __builtin_amdgcn_wmma_f32_16x16x4_f32(false, a, false, b, (short)0, acc, false, false);
    }
#pragma unroll
    for (int k = 0; k < 4; ++k) {   // A2 = T' x Hm,  T'[r][r'] = (r' > r)
        int kr = 4 * k + (half ? 2 : 0);
        v2f a; a.x = (kr > lm) ? 1.f : 0.f; a.y = (kr + 1 > lm) ? 1.f : 0.f;
        v2f b; b.x = hf[kr * 16 + lm]; b.y = hf[(kr + 1) * 16 + lm];
        acc2 = __builtin_amdgcn_wmma_f32_16x16x4_f32(false, a, false, b, (short)0, acc2, false, false);
    }
#pragma unroll
    for (int v = 0; v < 8; ++v) a2f[(v + (half ? 8 : 0)) * 16 + lm] = acc2[v];
    __syncthreads();
#pragma unroll
    for (int k = 0; k < 4; ++k) {   // acc += A2 x Ones
        int kr = 4 * k + (half ? 2 : 0);
        v2f a; a.x = a2f[lm * 16 + kr]; a.y = a2f[lm * 16 + kr + 1];
        v2f b; b.x = 1.f; b.y = 1.f;
        acc = __builtin_amdgcn_wmma_f32_16x16x4_f32(false, a, false, b, (short)0, acc, false, false);
    }
#pragma unroll
    for (int v = 0; v < 8; ++v) abv[(v + (half ? 8 : 0)) * 16 + lm] = acc[v];
#else
    if (lane == 0) {
        float run = 0.f;
        for (int b = 255; b >= 0; --b) { abv[b] = run; run += hf[b]; }
    }
#endif
    __syncthreads();

    if (lane == 0) {
        uint R   = (LVL == 0) ? (uint)TOPK : W[REMW + n];
        uint ctx = (LVL == 0) ? 0u : W[CTXW + n];
        if (R < 1u) R = 1u;
        int picked = -1; uint A = 0;
        for (int b = 255; b >= 0; --b) {
            uint Ab = (uint)(abv[b] + 0.5f);
            if (Ab < R && Ab + hu[b] >= R) { picked = b; A = Ab; break; }
        }
        if (picked < 0) { picked = 0; A = (uint)(abv[0] + 0.5f); }  // fewer than R total
        W[CTXW + n] = (ctx << 8) | (uint)picked;
        W[REMW + n] = (R > A) ? (R - A) : 1u;
    }
}

// ---- bitonic sort of candidates (key desc, index asc), emit sorted top-1000 ----
__global__ __launch_bounds__(1024) void sortK(uint* __restrict__ W) {
    __shared__ uint sk[CAP];
    __shared__ uint si[CAP];
    const int n = blockIdx.x;
    const int tid = threadIdx.x;
    uint cnt = W[CNTW + n];
    if (cnt > CAP) cnt = CAP;
    const uint2* cd = (const uint2*)(W + CANDW) + (size_t)n * CAP;
    for (int i = tid; i < CAP; i += 1024) {
        if (i < (int)cnt) { uint2 v = cd[i]; sk[i] = v.x; si[i] = v.y; }
        else              { sk[i] = 0u; si[i] = 0xFFFFFFFFu; }
    }
    __syncthreads();
    for (int k = 2; k <= CAP; k <<= 1) {
        for (int j = k >> 1; j > 0; j >>= 1) {
            for (int i = tid; i < CAP; i += 1024) {
                int x = i ^ j;
                if (x > i) {
                    uint ka = sk[i], kb = sk[x], ia = si[i], ib = si[x];
                    bool aFirst = (ka > kb) || (ka == kb && ia < ib);
                    bool desc   = ((i & k) == 0);
                    if (desc != aFirst) { sk[i] = kb; sk[x] = ka; si[i] = ib; si[x] = ia; }
                }
            }
            __syncthreads();
        }
    }
    for (int i = tid; i < TOPK; i += 1024) {
        W[TKEYW + n * TOPK + i] = sk[i];
        W[TIDXW + n * TOPK + i] = si[i];
    }
}

// ---- decode boxes, clip, class-offset boxes for NMS ----
__global__ __launch_bounds__(1024) void decodeK(const float* __restrict__ loc2,
                                                const float* __restrict__ reg,
                                                const int* __restrict__ isz,
                                                uint* __restrict__ W) {
    const int n = blockIdx.x;
    const int i = threadIdx.x;
    if (i >= TOPK) return;
    uint key = W[TKEYW + n * TOPK + i];
    uint e   = W[TIDXW + n * TOPK + i];
    float s  = __uint_as_float(key);
    bool valid = (s > 0.f) && (e < (uint)(HWN * NCLS));
    float b0 = 0.f, b1 = 0.f, b2 = 0.f, b3 = 0.f;
    uint lbl = 0;
    if (valid) {
        uint lidx = e / NCLS;
        uint c    = e - lidx * NCLS;
        lbl = c + 1;
        float lx = loc2[2 * lidx], ly = loc2[2 * lidx + 1];
        const float* rp = reg + (size_t)n * 4 * HWN;
        float r0 = rp[lidx], r1 = rp[HWN + lidx], r2 = rp[2 * HWN + lidx], r3 = rp[3 * HWN + lidx];
        float ih = (float)isz[2 * n], iw = (float)isz[2 * n + 1];
        b0 = fminf(fmaxf(lx - r0, 0.f), iw - 1.f);
        b1 = fminf(fmaxf(ly - r1, 0.f), ih - 1.f);
        b2 = fminf(fmaxf(lx + r2, 0.f), iw - 1.f);
        b3 = fminf(fmaxf(ly + r3, 0.f), ih - 1.f);
    }
    float off = (float)lbl * 100000.0f;
    float* bx = (float*)W + BOXW + ((size_t)n * TOPK + i) * 4;
    bx[0] = b0; bx[1] = b1; bx[2] = b2; bx[3] = b3;
    float* ob = (float*)W + OBOXW + ((size_t)n * TOPK + i) * 4;
    ob[0] = b0 + off; ob[1] = b1 + off; ob[2] = b2 + off; ob[3] = b3 + off;
    W[LBLW + n * TOPK + i] = lbl;
    W[VALW + n * TOPK + i] = valid ? 1u : 0u;
}

// ---- pairwise IoU adjacency bitmask (self bit cleared) ----
__global__ __launch_bounds__(256) void adjK(uint* __restrict__ W) {
    __shared__ float obx[TOPK * 4];
    __shared__ float ar[TOPK];
    const int n = blockIdx.y;
    const int tid = threadIdx.x;
    const float* ob = (const float*)W + OBOXW + (size_t)n * TOPK * 4;
    for (int i = tid; i < TOPK; i += 256) {
        float x1 = ob[i * 4], y1 = ob[i * 4 + 1], x2 = ob[i * 4 + 2], y2 = ob[i * 4 + 3];
        obx[i * 4] = x1; obx[i * 4 + 1] = y1; obx[i * 4 + 2] = x2; obx[i * 4 + 3] = y2;
        ar[i] = (x2 - x1 + 1.f) * (y2 - y1 + 1.f);
    }
    __syncthreads();
    const int row  = blockIdx.x * 8 + (tid >> 5);
    const int word = tid & 31;
    float x1 = obx[row * 4], y1 = obx[row * 4 + 1], x2 = obx[row * 4 + 2], y2 = obx[row * 4 + 3];
    float aR = ar[row];
    uint m = 0;
    const int jb = word * 32;
    for (int t = 0; t < 32; ++t) {
        int j = jb + t;
        if (j < TOPK && j != row) {
            float iw = fminf(x2, obx[j * 4 + 2]) - fmaxf(x1, obx[j * 4]) + 1.f;
            float ih = fminf(y2, obx[j * 4 + 3]) - fmaxf(y1, obx[j * 4 + 1]) + 1.f;
            iw = fmaxf(iw, 0.f); ih = fmaxf(ih, 0.f);
            float inter = iw * ih;
            float iou = inter / (aR + ar[j] - inter);
            if (iou > NMS_T) m |= (1u << t);
        }
    }
    W[ADJW + ((size_t)n * TOPK + row) * 32 + word] = m;
}

// ---- single-wave32 sequential NMS scan + output ----
__global__ __launch_bounds__(32) void nmsK(uint* __restrict__ W, float* __restrict__ out) {
    volatile __shared__ uint supp[32];
    volatile __shared__ uint keep[32];
    __shared__ uint vmask[32];
    const int n = blockIdx.x;
    const int lane = threadIdx.x;
    supp[lane] = 0u;
    keep[lane] = 0u;
    uint vw = 0;
    for (int t = 0; t < 32; ++t) {
        int i = lane * 32 + t;
        if (i < TOPK && W[VALW + n * TOPK + i]) vw |= (1u << t);
    }
    vmask[lane] = vw;
    __syncthreads();
    const uint* adjp = W + ADJW + (size_t)n * TOPK * 32;
    for (int i = 0; i < TOPK; ++i) {
        int wi = i >> 5, bi = i & 31;
        uint sb = (supp[wi] >> bi) & 1u;
        uint vb = (vmask[wi] >> bi) & 1u;
        if (vb && !sb) {
            if (lane == wi) keep[wi] = keep[wi] | (1u << bi);
            supp[lane] = supp[lane] | adjp[(size_t)i * 32 + lane];
        }
    }
    __syncthreads();
    if (lane == 0) {
        const float* bx = (const float*)W + BOXW + (size_t)n * TOPK * 4;
        int oc = 0;
        for (int i = 0; i < TOPK && oc < OUTN; ++i) {
            if ((keep[i >> 5] >> (i & 31)) & 1u) {
                float s  = __uint_as_float(W[TKEYW + n * TOPK + i]);
                float* o = out + ((size_t)n * OUTN + oc) * 6;
                o[0] = bx[i * 4]; o[1] = bx[i * 4 + 1]; o[2] = bx[i * 4 + 2]; o[3] = bx[i * 4 + 3];
                o[4] = sqrtf(s);
                o[5] = (float)W[LBLW + n * TOPK + i];
                ++oc;
            }
        }
        for (; oc < OUTN; ++oc) {
            float* o = out + ((size_t)n * OUTN + oc) * 6;
            for (int q = 0; q < 6; ++q) o[q] = 0.f;
        }
    }
}

extern "C" void kernel_launch(void* const* d_in, const int* in_sizes, int n_in,
                              void* d_out, int out_size, void* d_ws, size_t ws_size,
                              hipStream_t stream) {
    (void)in_sizes; (void)n_in; (void)out_size; (void)ws_size;
    const float* loc  = (const float*)d_in[0];
    const float* cls  = (const float*)d_in[1];
    const float* reg  = (const float*)d_in[2];
    const float* cent = (const float*)d_in[3];
    const int*   isz  = (const int*)d_in[4];
    uint*  W   = (uint*)d_ws;
    float* out = (float*)d_out;

    (void)hipMemsetAsync(d_ws, 0, ZERO_BYTES, stream);

    dim3 gs(NCHUNK, NIMG);
    scanK<0><<<gs, 256, 0, stream>>>(cls, cent, W);
    pickK<0><<<NIMG, 32, 0, stream>>>(W);
    scanK<1><<<gs, 256, 0, stream>>>(cls, cent, W);
    pickK<1><<<NIMG, 32, 0, stream>>>(W);
    scanK<2><<<gs, 256, 0, stream>>>(cls, cent, W);
    pickK<2><<<NIMG, 32, 0, stream>>>(W);
    scanK<3><<<gs, 256, 0, stream>>>(cls, cent, W);
    sortK<<<NIMG, 1024, 0, stream>>>(W);
    decodeK<<<NIMG, 1024, 0, stream>>>(loc, reg, isz, W);
    adjK<<<dim3(TOPK / 8, NIMG), 256, 0, stream>>>(W);
    nmsK<<<NIMG, 32, 0, stream>>>(W, out);
}